// Matching_Reducer_5712306504555
// MI455X (gfx1250) — compile-verified
//
#include <hip/hip_runtime.h>
#include <hip/hip_bf16.h>

typedef __attribute__((ext_vector_type(2))) float v2f;
typedef __attribute__((ext_vector_type(8))) float v8f;

#define NB   16
#define HIS  50
#define SIG  512
#define DIM  256
#define KTOP 8
#define SP   511          // scores length per (b,h)
#define ROWS_OUT 449      // H*9 - 1

#define PS_ELEMS   (NB * ROWS_OUT * DIM)   // 1,839,104
#define MASK_ELEMS (NB * ROWS_OUT)         // 7,184

__global__ __launch_bounds__(256)
void matching_reducer_kernel(
    const float* __restrict__ sel,     // (B,H,S,D)
    const float* __restrict__ emb,     // (B,H,S,D)
    const float* __restrict__ user,    // (B,1,D)
    const float* __restrict__ nrep,    // (B,H,D)
    const int*   __restrict__ attn,    // (B,H,S)
    const int*   __restrict__ refm,    // (B,H,S)
    const float* __restrict__ Wal,     // (D, 2D)
    const float* __restrict__ bal,     // (D,)
    const float* __restrict__ sepv,    // (D,)
    const float* __restrict__ ordemb,  // (H,1,D)
    float* __restrict__ out_ps,
    float* __restrict__ out_mask,
    float* __restrict__ out_kid)
{
    __shared__ float nu[2 * DIM];   // staged [user ; news_repr]
    __shared__ float qn[DIM];       // normalized query
    __shared__ float sc[512];       // raw -> masked scores (pad idx 511)
    __shared__ float n2[512];       // per-row squared norms
    __shared__ float red[8];
    __shared__ float red_v[8];
    __shared__ int   red_i[8];
    __shared__ int   tk_i[KTOP];
    __shared__ float tk_v[KTOP];
    __shared__ float tk_w[KTOP];

    const int tid  = threadIdx.x;          // 0..255
    const int lane = tid & 31;
    const int wv   = tid >> 5;             // 0..7
    const int bh   = blockIdx.x;           // 0..799
    const int b    = bh / HIS;
    const int h    = bh % HIS;

    // ---- Phase A: stage nu = [user ; news_repr] into LDS ----
    nu[tid]        = user[b * DIM + tid];
    nu[DIM + tid]  = nrep[(size_t)bh * DIM + tid];
    __syncthreads();

    // ---- Phase B: query[d] = b[d] + W[d,:] . nu ; then l2-normalize ----
    {
        const float* wrow = Wal + (size_t)tid * (2 * DIM);
        float acc = bal[tid];
        #pragma unroll 4
        for (int j = 0; j < 2 * DIM; ++j) acc = fmaf(wrow[j], nu[j], acc);

        float ss = acc * acc;
        #pragma unroll
        for (int o = 16; o > 0; o >>= 1) ss += __shfl_xor(ss, o, 32);
        if (lane == 0) red[wv] = ss;
        __syncthreads();
        if (tid == 0) {
            float t = 0.f;
            #pragma unroll
            for (int i = 0; i < 8; ++i) t += red[i];
            red[0] = t;
        }
        __syncthreads();
        const float nrm = sqrtf(red[0]);
        qn[tid] = acc / fmaxf(nrm, 1e-12f);
    }
    __syncthreads();

    // ---- Phase C: scores via V_WMMA_F32_16X16X4_F32 ----
    // 32 tiles of 16 rows cover s = 1..512 (row 512 is a clamped pad -> -inf later).
    // A: 16 sel rows x 4 k's. Lane layout: lanes 0-15 hold {k0,k0+1} of row r=lane,
    //    lanes 16-31 hold {k0+2,k0+3} of row r=lane-16  -> 8-byte loads per lane.
    // B: query chunk broadcast to all 16 columns (same half-lane k split).
    // D: all 16 columns identical; lane 0 holds tile rows 0..7, lane 16 rows 8..15.
    {
        const float* base = sel + (size_t)bh * SIG * DIM;
        const int r    = lane & 15;
        const int koff = (lane >> 4) << 1;           // 0 or 2

        for (int t = wv; t < 32; t += 8) {
            const int s      = 1 + t * 16 + r;       // 1..512
            const int sc_idx = t * 16 + r;           // 0..511
            const int s_cl   = (s < SIG) ? s : (SIG - 1);
            const float* rowp = base + (size_t)s_cl * DIM + koff;
            const float* qnp  = qn + koff;

            v8f c = {0.f, 0.f, 0.f, 0.f, 0.f, 0.f, 0.f, 0.f};
            float ssq = 0.f;
            #pragma unroll 8
            for (int k0 = 0; k0 < DIM; k0 += 4) {
                v2f a  = *(const v2f*)(rowp + k0);   // global b64 load
                v2f bq = *(const v2f*)(qnp + k0);    // ds b64 load
                c = __builtin_amdgcn_wmma_f32_16x16x4_f32(
                        false, a, false, bq, (short)0, c, false, false);
                ssq = fmaf(a.x, a.x, ssq);
                ssq = fmaf(a.y, a.y, ssq);
            }
            // full row norm^2: combine the two half-lane partial sums
            const float nn = ssq + __shfl_xor(ssq, 16, 32);
            if (lane < 16) n2[sc_idx] = nn;
            if (lane == 0) {
                #pragma unroll
                for (int j = 0; j < 8; ++j) sc[t * 16 + j] = c[j];
            } else if (lane == 16) {
                #pragma unroll
                for (int j = 0; j < 8; ++j) sc[t * 16 + 8 + j] = c[j];
            }
        }
    }
    __syncthreads();

    // ---- Phase D: normalize by row norm + apply pad mask ----
    for (int j = tid; j < 512; j += 256) {
        float v;
        if (j < SP) {
            v = sc[j] / fmaxf(sqrtf(n2[j]), 1e-12f);
            // pad iff (refined_mask[s] + keep_first_9[s]) == 0 with s = j+1
            if (j >= KTOP && refm[(size_t)bh * SIG + (j + 1)] == 0)
                v = -__builtin_inff();
        } else {
            v = -__builtin_inff();
        }
        sc[j] = v;
    }
    __syncthreads();

    // ---- Phase E: top-8 (descending, ties -> smaller index) ----
    for (int it = 0; it < KTOP; ++it) {
        float bv = -__builtin_inff();
        int   bi = 0x7fffffff;
        for (int j = tid; j < 512; j += 256) {
            const float v = sc[j];
            if (v > bv || (v == bv && j < bi)) { bv = v; bi = j; }
        }
        #pragma unroll
        for (int o = 16; o > 0; o >>= 1) {
            const float ov = __shfl_xor(bv, o, 32);
            const int   oi = __shfl_xor(bi, o, 32);
            if (ov > bv || (ov == bv && oi < bi)) { bv = ov; bi = oi; }
        }
        if (lane == 0) { red_v[wv] = bv; red_i[wv] = bi; }
        __syncthreads();
        if (tid == 0) {
            float fv = red_v[0]; int fi = red_i[0];
            for (int i = 1; i < 8; ++i) {
                if (red_v[i] > fv || (red_v[i] == fv && red_i[i] < fi)) {
                    fv = red_v[i]; fi = red_i[i];
                }
            }
            tk_v[it] = fv; tk_i[it] = fi;
            sc[fi] = -__builtin_inff();
        }
        __syncthreads();
    }

    // ---- Phase F: softmax over the 8 selected scores ----
    if (tid == 0) {
        float m = tk_v[0];
        #pragma unroll
        for (int t = 1; t < KTOP; ++t) m = fmaxf(m, tk_v[t]);
        float s = 0.f; float e[KTOP];
        #pragma unroll
        for (int t = 0; t < KTOP; ++t) { e[t] = __expf(tk_v[t] - m); s += e[t]; }
        #pragma unroll
        for (int t = 0; t < KTOP; ++t) tk_w[t] = e[t] / s;
    }
    __syncthreads();

    // ---- Phase G: outputs ----
    if (tid < KTOP)
        out_kid[(size_t)bh * KTOP + tid] = (float)tk_i[tid];

    if (tid < 9) {
        const int g = h * 9 + tid;
        if (g < ROWS_OUT) {
            float mv = (tid < KTOP)
                ? (float)attn[(size_t)bh * SIG + tk_i[tid] + 1]
                : 1.f;
            out_mask[(size_t)b * ROWS_OUT + g] = mv;
        }
    }

    const float* embb = emb + (size_t)bh * SIG * DIM;
    const float* oeb  = ordemb + (size_t)h * DIM;
    for (int t = 0; t < 9; ++t) {
        const int g = h * 9 + t;
        if (g >= ROWS_OUT) break;
        float v;
        if (t < KTOP) {
            v = fmaf(embb[(size_t)(tk_i[t] + 1) * DIM + tid], tk_w[t], oeb[tid]);
        } else {
            v = sepv[tid];
        }
        out_ps[((size_t)b * ROWS_OUT + g) * DIM + tid] = v;
    }
}

extern "C" void kernel_launch(void* const* d_in, const int* in_sizes, int n_in,
                              void* d_out, int out_size, void* d_ws, size_t ws_size,
                              hipStream_t stream) {
    (void)in_sizes; (void)n_in; (void)out_size; (void)d_ws; (void)ws_size;

    const float* sel    = (const float*)d_in[0];
    const float* emb    = (const float*)d_in[1];
    const float* user   = (const float*)d_in[2];
    const float* nrep   = (const float*)d_in[3];
    const int*   attn   = (const int*)  d_in[4];
    const int*   refm   = (const int*)  d_in[5];
    const float* Wal    = (const float*)d_in[6];
    const float* bal    = (const float*)d_in[7];
    const float* sepv   = (const float*)d_in[8];
    const float* ordemb = (const float*)d_in[9];

    float* out_ps   = (float*)d_out;
    float* out_mask = out_ps + PS_ELEMS;
    float* out_kid  = out_mask + MASK_ELEMS;

    matching_reducer_kernel<<<dim3(NB * HIS), dim3(256), 0, stream>>>(
        sel, emb, user, nrep, attn, refm, Wal, bal, sepv, ordemb,
        out_ps, out_mask, out_kid);
}